// Level0SSM_41652592836928
// MI455X (gfx1250) — compile-verified
//
#include <hip/hip_runtime.h>
#include <hip/hip_bf16.h>
#include <math.h>

// ---------------------------------------------------------------------------
// Level0SSM for MI455X (gfx1250): selective-SSM stack, fp32 WMMA 16x16x4.
// One wave32x16 workgroup per sequence; LDS-resident intermediates; register
// scan state (16 states / lane).
// ---------------------------------------------------------------------------

typedef float v2f __attribute__((ext_vector_type(2)));
typedef float v8f __attribute__((ext_vector_type(8)));

#define BSZc 8
#define NOBJ 64
#define TT   128
#define DD   256
#define DIc  512
#define DSc  16
#define TC   64      // T chunk held in LDS
#define NWV  16      // waves per workgroup (512 threads)

// --- WMMA f32 16x16x4 fragment helpers -------------------------------------
// A 16x4 layout: lanes0-15 -> M=lane, K=k0+{0,1}; lanes16-31 -> M=lane-16, K=k0+{2,3}
// B  4x16 layout (as B^T like A with N in place of M)
// C/D 16x16: element(m,n): m = v + 8*(lane>>4), n = lane&15  (v = vgpr index)

static __device__ __forceinline__ v2f frag_a_f32(const float* p, int m0, int k0,
                                                 int ld, int lane) {
  int m = m0 + (lane & 15);
  int k = k0 + ((lane >> 4) << 1);
  v2f a; a[0] = p[m * ld + k]; a[1] = p[m * ld + k + 1]; return a;
}
static __device__ __forceinline__ v2f frag_a_f16(const _Float16* p, int m0, int k0,
                                                 int ld, int lane) {
  int m = m0 + (lane & 15);
  int k = k0 + ((lane >> 4) << 1);
  v2f a; a[0] = (float)p[m * ld + k]; a[1] = (float)p[m * ld + k + 1]; return a;
}
// weights stored row-major W[N_total][K]; B[k][n] = W[n0+n][k]
static __device__ __forceinline__ v2f frag_b_w(const float* W, int n0, int k0,
                                               int K, int lane) {
  int n = n0 + (lane & 15);
  int k = k0 + ((lane >> 4) << 1);
  const float* q = W + (size_t)n * K + k;
  v2f b; b[0] = q[0]; b[1] = q[1]; return b;
}

#define WMMA4(acc, a, b)                                                       \
  acc = __builtin_amdgcn_wmma_f32_16x16x4_f32(false, (a), false, (b),          \
                                              (short)0, (acc), false, false)

// ---------------------------------------------------------------------------
// Fused SSM block: x (S,T,D) updated in place: x += gate*scan(...) @ out_w^T
// ---------------------------------------------------------------------------
__global__ __launch_bounds__(512, 1)
void ssm_block_kernel(float* __restrict__ x, int S,
                      _Float16* __restrict__ gate_ws,
                      const float* __restrict__ norm_w,
                      const float* __restrict__ norm_b,
                      const float* __restrict__ in_w,
                      const float* __restrict__ gate_w,
                      const float* __restrict__ dt_w,
                      const float* __restrict__ dt_b,
                      const float* __restrict__ Bw,
                      const float* __restrict__ Cw,
                      const float* __restrict__ A_log,
                      const float* __restrict__ out_w) {
  __shared__ float     s_r0[TC * DD];   // 64KB: xn (f32); aliased as dt (f16 TCxDI)
  __shared__ _Float16  s_z[TC * DIc];   // 64KB: z (f16) -> overwritten with ys*gate
  __shared__ float     s_B[TC * DSc];   // 4KB
  __shared__ float     s_C[TC * DSc];   // 4KB
  _Float16* s_dt = (_Float16*)s_r0;

  const int seq  = blockIdx.x;
  const int tid  = threadIdx.x;
  const int w    = tid >> 5;
  const int lane = tid & 31;
  (void)S;

  const v8f vzero = {0.f, 0.f, 0.f, 0.f, 0.f, 0.f, 0.f, 0.f};

  // scan state: thread j owns channel j (DI == blockDim.x)
  float av[DSc], h[DSc];
#pragma unroll
  for (int s = 0; s < DSc; ++s) {
    float e = __expf(A_log[tid * DSc + s]);
    av[s] = -fminf(fmaxf(e, 1e-6f), 1.0f);
    h[s] = 0.f;
  }

  for (int c = 0; c < TT / TC; ++c) {
    const int t0 = c * TC;

    // ---- Phase A: LayerNorm rows -> s_r0 (one wave per row) ----
    for (int i = 0; i < TC / NWV; ++i) {
      const int r = w + NWV * i;
      const float* xr = x + ((size_t)seq * TT + t0 + r) * DD;
      const int c0 = lane * 8;
      float xv[8];
#pragma unroll
      for (int u = 0; u < 8; ++u) xv[u] = xr[c0 + u];
      float sm = 0.f;
#pragma unroll
      for (int u = 0; u < 8; ++u) sm += xv[u];
#pragma unroll
      for (int off = 16; off > 0; off >>= 1) sm += __shfl_xor(sm, off, 32);
      const float mean = sm * (1.0f / DD);
      float vs = 0.f;
#pragma unroll
      for (int u = 0; u < 8; ++u) { float d = xv[u] - mean; vs += d * d; }
#pragma unroll
      for (int off = 16; off > 0; off >>= 1) vs += __shfl_xor(vs, off, 32);
      const float rs = rsqrtf(vs * (1.0f / DD) + 1e-5f);
#pragma unroll
      for (int u = 0; u < 8; ++u)
        s_r0[r * DD + c0 + u] =
            (xv[u] - mean) * rs * norm_w[c0 + u] + norm_b[c0 + u];
    }
    __syncthreads();

    // ---- Phase B: z = xn@in_w^T (->LDS f16); gate = silu(xn@gate_w^T) ----
    for (int jb = w; jb < 64; jb += NWV) {
      const int mat = jb >> 5;            // 0: z, 1: gate
      const int n0  = (jb & 31) * 16;
      const float* W = mat ? gate_w : in_w;
      v8f acc[4];
#pragma unroll
      for (int mt = 0; mt < 4; ++mt) acc[mt] = vzero;
      for (int k0 = 0; k0 < DD; k0 += 4) {
        v2f bf = frag_b_w(W, n0, k0, DD, lane);
#pragma unroll
        for (int mt = 0; mt < 4; ++mt) {
          v2f af = frag_a_f32(s_r0, mt * 16, k0, DD, lane);
          WMMA4(acc[mt], af, bf);
        }
      }
      const int ncol = n0 + (lane & 15);
      const int mh   = (lane >> 4) * 8;
#pragma unroll
      for (int mt = 0; mt < 4; ++mt) {
#pragma unroll
        for (int v = 0; v < 8; ++v) {
          const int m = mt * 16 + v + mh;
          const float val = acc[mt][v];
          if (mat == 0) {
            s_z[m * DIc + ncol] = (_Float16)val;
          } else {
            const float sg = val / (1.0f + __expf(-val));
            gate_ws[((size_t)seq * TT + t0 + m) * DIc + ncol] = (_Float16)sg;
          }
        }
      }
    }
    __threadfence();     // gate_ws stores visible before Phase D reads
    __syncthreads();

    // ---- Phase C: dt = clip(softplus(z@dt_w^T+b)); Bp/Cp = z@{B,C}_w^T ----
    for (int jb = w; jb < 34; jb += NWV) {
      const float* W; int n0, kind;
      if (jb < 32)      { W = dt_w; n0 = jb * 16; kind = 0; }
      else if (jb == 32){ W = Bw;   n0 = 0;       kind = 1; }
      else              { W = Cw;   n0 = 0;       kind = 2; }
      v8f acc[4];
#pragma unroll
      for (int mt = 0; mt < 4; ++mt) acc[mt] = vzero;
      for (int k0 = 0; k0 < DIc; k0 += 4) {
        v2f bf = frag_b_w(W, n0, k0, DIc, lane);
#pragma unroll
        for (int mt = 0; mt < 4; ++mt) {
          v2f af = frag_a_f16(s_z, mt * 16, k0, DIc, lane);
          WMMA4(acc[mt], af, bf);
        }
      }
      const int nlo = lane & 15;
      const int mh  = (lane >> 4) * 8;
#pragma unroll
      for (int mt = 0; mt < 4; ++mt) {
#pragma unroll
        for (int v = 0; v < 8; ++v) {
          const int m = mt * 16 + v + mh;
          const float val = acc[mt][v];
          if (kind == 0) {
            const int colc = n0 + nlo;
            const float zz = val + dt_b[colc];
            float sp = fmaxf(zz, 0.f) + log1pf(__expf(-fabsf(zz)));
            sp = fminf(fmaxf(sp, 1e-3f), 1.0f);
            s_dt[m * DIc + colc] = (_Float16)sp;
          } else if (kind == 1) {
            s_B[m * DSc + nlo] = val;
          } else {
            s_C[m * DSc + nlo] = val;
          }
        }
      }
    }
    __syncthreads();

    // ---- Phase D: sequential scan, barrier-free (thread j == channel j) ----
    {
      const int j = tid;
      for (int t = 0; t < TC; ++t) {
        const float zt  = (float)s_z[t * DIc + j];
        const float dtt = (float)s_dt[t * DIc + j];
        float y = 0.f;
#pragma unroll
        for (int s = 0; s < DSc; ++s) {
          const float dA = __expf(fminf(fmaxf(dtt * av[s], -20.f), 0.f));
          h[s] = dA * h[s] + (dtt * s_B[t * DSc + s]) * zt;
          y += h[s] * s_C[t * DSc + s];
        }
        const float gt =
            (float)gate_ws[((size_t)seq * TT + t0 + t) * DIc + j];
        s_z[t * DIc + j] = (_Float16)(y * gt);   // ys*gate overwrites z row
      }
    }
    __syncthreads();

    // ---- Phase E: x += ys @ out_w^T (residual RMW into global) ----
    {
      const int n0 = w * 16;
      v8f acc[4];
#pragma unroll
      for (int mt = 0; mt < 4; ++mt) acc[mt] = vzero;
      for (int k0 = 0; k0 < DIc; k0 += 4) {
        v2f bf = frag_b_w(out_w, n0, k0, DIc, lane);
#pragma unroll
        for (int mt = 0; mt < 4; ++mt) {
          v2f af = frag_a_f16(s_z, mt * 16, k0, DIc, lane);
          WMMA4(acc[mt], af, bf);
        }
      }
      const int ncol = n0 + (lane & 15);
      const int mh   = (lane >> 4) * 8;
#pragma unroll
      for (int mt = 0; mt < 4; ++mt) {
#pragma unroll
        for (int v = 0; v < 8; ++v) {
          const int m = mt * 16 + v + mh;
          const size_t xi = ((size_t)seq * TT + t0 + m) * DD + ncol;
          x[xi] = x[xi] + acc[mt][v];
        }
      }
    }
    __syncthreads();
  }
}

// ---------------------------------------------------------------------------
// Small helper kernels
// ---------------------------------------------------------------------------
__global__ void pool_kernel(const float* __restrict__ obj,
                            const float* __restrict__ mask,
                            float* __restrict__ g) {
  const int bt = blockIdx.x;           // b*T + t
  const int b  = bt >> 7;
  const int t  = bt & (TT - 1);
  const int d  = threadIdx.x;
  float na = 0.f;
  for (int n = 0; n < NOBJ; ++n) na += mask[b * NOBJ + n];
  na = fmaxf(na, 1.0f);
  float acc = 0.f;
  for (int n = 0; n < NOBJ; ++n) {
    const float mk = mask[b * NOBJ + n];
    acc += mk * obj[(((size_t)(b * NOBJ + n)) * TT + t) * DD + d];
  }
  g[(size_t)bt * DD + d] = acc / na;
}

__global__ void cond_kernel(const float* __restrict__ g,
                            const float* __restrict__ cw,
                            float* __restrict__ cond) {
  __shared__ float row[DD];
  const int bt = blockIdx.x;
  const int d  = threadIdx.x;
  row[d] = g[(size_t)bt * DD + d];
  __syncthreads();
  float acc = 0.f;
  for (int k = 0; k < DD; ++k) acc += cw[(size_t)d * DD + k] * row[k];
  cond[(size_t)bt * DD + d] = acc;
}

__global__ void add_kernel(const float* __restrict__ obj,
                           const float* __restrict__ cond,
                           float* __restrict__ xo) {
  const size_t idx = (size_t)blockIdx.x * blockDim.x + threadIdx.x;
  const int d  = (int)(idx & (DD - 1));
  const int t  = (int)((idx >> 8) & (TT - 1));
  const int bn = (int)(idx >> 15);
  const int b  = bn >> 6;
  xo[idx] = obj[idx] + cond[(((size_t)(b << 7) + t) << 8) + d];
}

__global__ void finalize_kernel(float* __restrict__ xo,
                                const float* __restrict__ mask) {
  const size_t idx = (size_t)blockIdx.x * blockDim.x + threadIdx.x;
  const int bn = (int)(idx >> 15);
  float v = xo[idx];
  if (v != v) v = 0.f;                        // nan -> 0
  v = fminf(fmaxf(v, -100.f), 100.f);         // +/-inf & clip -> +/-100
  xo[idx] = v * mask[bn];
}

// ---------------------------------------------------------------------------
extern "C" void kernel_launch(void* const* d_in, const int* in_sizes, int n_in,
                              void* d_out, int out_size, void* d_ws,
                              size_t ws_size, hipStream_t stream) {
  (void)in_sizes; (void)n_in; (void)out_size; (void)ws_size;
  const float* obj    = (const float*)d_in[0];
  const float* mask   = (const float*)d_in[1];
  const float* cond_w = (const float*)d_in[2];
  const float* gp[10];
  const float* op[10];
  for (int i = 0; i < 10; ++i) gp[i] = (const float*)d_in[3 + i];
  for (int i = 0; i < 10; ++i) op[i] = (const float*)d_in[13 + i];

  // workspace carve: gate (f16, 512*128*512) | g (f32) | cond (f32)
  char* ws = (char*)d_ws;
  _Float16* gate_ws = (_Float16*)ws;
  size_t off = (size_t)512 * TT * DIc * sizeof(_Float16);
  float* g_ws = (float*)(ws + off);
  off += (size_t)BSZc * TT * DD * sizeof(float);
  float* cond_ws = (float*)(ws + off);

  float* xout = (float*)d_out;      // object-track x lives in d_out

  // 1) masked mean-pool -> g
  pool_kernel<<<dim3(BSZc * TT), dim3(DD), 0, stream>>>(obj, mask, g_ws);

  // 2) global-track SSM blocks
  for (int i = 0; i < 2; ++i) {
    ssm_block_kernel<<<dim3(BSZc), dim3(512), 0, stream>>>(
        g_ws, BSZc, gate_ws,
        gp[0] + (size_t)i * DD, gp[1] + (size_t)i * DD,
        gp[2] + (size_t)i * DIc * DD, gp[3] + (size_t)i * DIc * DD,
        gp[4] + (size_t)i * DIc * DIc, gp[5] + (size_t)i * DIc,
        gp[6] + (size_t)i * DSc * DIc, gp[7] + (size_t)i * DSc * DIc,
        gp[8] + (size_t)i * DIc * DSc, gp[9] + (size_t)i * DD * DIc);
  }

  // 3) cond = g @ cond_w^T ; x = obj + cond (broadcast over N) -> d_out
  cond_kernel<<<dim3(BSZc * TT), dim3(DD), 0, stream>>>(g_ws, cond_w, cond_ws);
  const size_t total = (size_t)BSZc * NOBJ * TT * DD;
  add_kernel<<<dim3((unsigned)(total / 512)), dim3(512), 0, stream>>>(
      obj, cond_ws, xout);

  // 4) object-track SSM blocks (512 sequences)
  for (int i = 0; i < 2; ++i) {
    ssm_block_kernel<<<dim3(BSZc * NOBJ), dim3(512), 0, stream>>>(
        xout, BSZc * NOBJ, gate_ws,
        op[0] + (size_t)i * DD, op[1] + (size_t)i * DD,
        op[2] + (size_t)i * DIc * DD, op[3] + (size_t)i * DIc * DD,
        op[4] + (size_t)i * DIc * DIc, op[5] + (size_t)i * DIc,
        op[6] + (size_t)i * DSc * DIc, op[7] + (size_t)i * DSc * DIc,
        op[8] + (size_t)i * DIc * DSc, op[9] + (size_t)i * DD * DIc);
  }

  // 5) nan_to_num + clip + mask
  finalize_kernel<<<dim3((unsigned)(total / 512)), dim3(512), 0, stream>>>(
      xout, mask);
}